// TransactionGNN_7310034338256
// MI455X (gfx1250) — compile-verified
//
#include <hip/hip_runtime.h>
#include <cstddef>

// ---------------------------------------------------------------------------
// GCN 3-layer forward for MI455X (gfx1250, wave32).
// Dense GEMMs use V_WMMA_F32_16X16X4_F32 (fp32-precise matrix pipe).
// Edge aggregation uses coalesced float4 gathers + global_atomic_add_f32.
// ---------------------------------------------------------------------------

typedef float v2f __attribute__((ext_vector_type(2)));
typedef float v8f __attribute__((ext_vector_type(8)));

#define GCN_N_NODES  100000
#define GCN_N_GRAPHS 512

__device__ __forceinline__ void atom_add_f32(float* p, float v) {
    // non-returning relaxed agent-scope fp32 atomic -> global_atomic_add_f32
    __hip_atomic_fetch_add(p, v, __ATOMIC_RELAXED, __HIP_MEMORY_SCOPE_AGENT);
}

// ---------------------------------------------------------------------------
// Dense GEMM  H[N,DOUT] = act(X[N,DIN]) @ W[DIN,DOUT]
// One wave32 per 16x16 output tile; 8 waves (= 8 column tiles for DOUT=128)
// per block share the same 16 rows of X -> L0/L2 reuse.
// ---------------------------------------------------------------------------
template <int DIN, int DOUT, bool RELU>
__global__ __launch_bounds__(256)
void gemm_wmma_f32(const float* __restrict__ X,
                   const float* __restrict__ W,
                   float* __restrict__ H) {
    const int lane = threadIdx.x & 31;
    const int wave = threadIdx.x >> 5;
    const int colTiles = DOUT / 16;

    const int tile    = blockIdx.x * 8 + wave;
    const int rowTile = tile / colTiles;
    const int colTile = tile % colTiles;
    const int row0 = rowTile * 16;
    const int col0 = colTile * 16;

    const int m  = lane & 15;            // M index within tile
    const int kb = (lane >> 4) << 1;     // K sub-offset: 0 (lanes 0-15) or 2 (16-31)

    const float* __restrict__ xrow = X + (size_t)(row0 + m) * DIN;
    const float* __restrict__ wcol = W + col0 + m;   // N index == lane&15

    v8f acc = {};
#pragma unroll
    for (int k0 = 0; k0 < DIN; k0 += 4) {
        v2f a;
        a.x = xrow[k0 + kb + 0];
        a.y = xrow[k0 + kb + 1];
        if (RELU) { a.x = fmaxf(a.x, 0.0f); a.y = fmaxf(a.y, 0.0f); }
        v2f b;
        b.x = wcol[(size_t)(k0 + kb + 0) * DOUT];
        b.y = wcol[(size_t)(k0 + kb + 1) * DOUT];
        acc = __builtin_amdgcn_wmma_f32_16x16x4_f32(
            /*neg_a=*/false, a, /*neg_b=*/false, b,
            /*c_mod=*/(short)0, acc, /*reuse_a=*/false, /*reuse_b=*/false);
    }

    // C/D layout: VGPR r -> M=r (lanes 0-15) / M=r+8 (lanes 16-31), N=lane&15
    const int colw  = col0 + m;
    const int rbase = row0 + ((lane >> 4) << 3);
#pragma unroll
    for (int r = 0; r < 8; ++r)
        H[(size_t)(rbase + r) * DOUT + colw] = acc[r];
}

// ---------------------------------------------------------------------------
// Small helper kernels
// ---------------------------------------------------------------------------
__global__ void fill_f32(float* __restrict__ p, float v, int n) {
    int i = blockIdx.x * blockDim.x + threadIdx.x;
    if (i < n) p[i] = v;
}

__global__ void deg_accum(const int* __restrict__ dst, float* __restrict__ deg, int e) {
    int i = blockIdx.x * blockDim.x + threadIdx.x;
    if (i < e) atom_add_f32(&deg[dst[i]], 1.0f);
}

__global__ void rsqrt_inplace(float* __restrict__ p, int n) {
    int i = blockIdx.x * blockDim.x + threadIdx.x;
    if (i < n) p[i] = rsqrtf(p[i]);
}

// AGG[i,f] = H[i,f] * dinv[i]^2 + bias[f]   (self-loop contribution + bias)
template <int D>
__global__ void self_bias(const float* __restrict__ H, const float* __restrict__ dinv,
                          const float* __restrict__ bias, float* __restrict__ AGG, int n) {
    const int CH = D / 4;
    int t = blockIdx.x * blockDim.x + threadIdx.x;
    if (t >= n * CH) return;
    int i = t / CH, c = t % CH;
    float di = dinv[i];
    float s  = di * di;
    float4 h = ((const float4*)(H + (size_t)i * D))[c];
    float4 b = ((const float4*)bias)[c];
    float4 o;
    o.x = h.x * s + b.x;  o.y = h.y * s + b.y;
    o.z = h.z * s + b.z;  o.w = h.w * s + b.w;
    ((float4*)(AGG + (size_t)i * D))[c] = o;
}

// AGG[dst] += H[src] * (dinv[src]*dinv[dst]) over all edges.
// Lane c handles float4 chunk c of the row -> coalesced 128B gathers per wave.
template <int D>
__global__ void scatter_edges(const int* __restrict__ src, const int* __restrict__ dst,
                              const float* __restrict__ dinv,
                              const float* __restrict__ H, float* __restrict__ AGG,
                              int e) {
    const int CH = D / 4;
    long long t = (long long)blockIdx.x * blockDim.x + threadIdx.x;
    if (t >= (long long)e * CH) return;
    int eid = (int)(t / CH), c = (int)(t % CH);
    int s = src[eid], d = dst[eid];
    float nm = dinv[s] * dinv[d];
    float4 h = ((const float4*)(H + (size_t)s * D))[c];
    float* out = AGG + (size_t)d * D + (size_t)c * 4;
    atom_add_f32(out + 0, h.x * nm);
    atom_add_f32(out + 1, h.y * nm);
    atom_add_f32(out + 2, h.z * nm);
    atom_add_f32(out + 3, h.w * nm);
}

// batch is sorted: accumulate runs in registers, flush one atomic per
// (graph, feature) per segment boundary. blockDim = 64 (feature = thread).
__global__ __launch_bounds__(64)
void pool_sum(const float* __restrict__ H, const int* __restrict__ batch,
              float* __restrict__ sums, float* __restrict__ cnt, int n) {
    const int NODES = 256;
    const int f  = threadIdx.x;       // 0..63
    const int i0 = blockIdx.x * NODES;
    if (i0 >= n) return;
    const int i1 = (i0 + NODES < n) ? (i0 + NODES) : n;

    int   curg = batch[i0];
    float acc = 0.0f, c = 0.0f;
    for (int i = i0; i < i1; ++i) {
        int g = batch[i];
        if (g != curg) {
            atom_add_f32(&sums[(size_t)curg * 64 + f], acc);
            if (f == 0) atom_add_f32(&cnt[curg], c);
            acc = 0.0f; c = 0.0f; curg = g;
        }
        acc += H[(size_t)i * 64 + f];
        c   += 1.0f;
    }
    atom_add_f32(&sums[(size_t)curg * 64 + f], acc);
    if (f == 0) atom_add_f32(&cnt[curg], c);
}

__global__ void pool_div(const float* __restrict__ sums, const float* __restrict__ cnt,
                         float* __restrict__ out, int total) {
    int t = blockIdx.x * blockDim.x + threadIdx.x;
    if (t >= total) return;
    int g = t >> 6;
    out[t] = sums[t] / fmaxf(cnt[g], 1.0f);
}

// ---------------------------------------------------------------------------
// Orchestration
// ---------------------------------------------------------------------------
extern "C" void kernel_launch(void* const* d_in, const int* in_sizes, int n_in,
                              void* d_out, int out_size, void* d_ws, size_t ws_size,
                              hipStream_t stream) {
    (void)n_in; (void)ws_size;

    const float* x  = (const float*)d_in[0];
    const float* W1 = (const float*)d_in[1];
    const float* b1 = (const float*)d_in[2];
    const float* W2 = (const float*)d_in[3];
    const float* b2 = (const float*)d_in[4];
    const float* W3 = (const float*)d_in[5];
    const float* b3 = (const float*)d_in[6];
    const int*   ei = (const int*)d_in[7];   // [2, E] flat
    const int*   bt = (const int*)d_in[8];   // [N], sorted

    const int n = in_sizes[0] / 128;         // 100000
    const int E = in_sizes[7] / 2;           // 1600000
    const int* src = ei;
    const int* dst = ei + E;

    // workspace layout (floats)
    float* ws   = (float*)d_ws;
    float* dinv = ws;                               // N
    float* bufA = dinv + n;                         // N*128 (GEMM outputs h)
    float* bufB = bufA + (size_t)n * 128;           // N*128 (aggregation)
    float* sums = bufB + (size_t)n * 128;           // 512*64
    float* cnt  = sums + (size_t)GCN_N_GRAPHS * 64; // 512
    float* outf = (float*)d_out;                    // 512*64

    const int T = 256;
    auto cdiv = [](long long a, long long b) { return (int)((a + b - 1) / b); };

    // --- degree -> dinv = rsqrt(1 + indeg) (shared by all 3 layers) ---------
    fill_f32<<<cdiv(n, T), T, 0, stream>>>(dinv, 1.0f, n);
    deg_accum<<<cdiv(E, T), T, 0, stream>>>(dst, dinv, E);
    rsqrt_inplace<<<cdiv(n, T), T, 0, stream>>>(dinv, n);

    const int rowTiles = n / 16;                       // 6250 (exact)
    const int blk128   = rowTiles * (128 / 16) / 8;    // 6250 blocks, 8 waves each
    const int blk64    = rowTiles * (64 / 16) / 8;     // 3125 blocks

    // --- layer 1: h1 = x @ W1 ; agg1 = scatter + self + b1 ------------------
    gemm_wmma_f32<128, 128, false><<<blk128, 256, 0, stream>>>(x, W1, bufA);
    self_bias<128><<<cdiv((long long)n * 32, T), T, 0, stream>>>(bufA, dinv, b1, bufB, n);
    scatter_edges<128><<<cdiv((long long)E * 32, T), T, 0, stream>>>(src, dst, dinv, bufA, bufB, E);

    // --- layer 2: h2 = relu(agg1) @ W2 ; agg2 -------------------------------
    gemm_wmma_f32<128, 128, true><<<blk128, 256, 0, stream>>>(bufB, W2, bufA);
    self_bias<128><<<cdiv((long long)n * 32, T), T, 0, stream>>>(bufA, dinv, b2, bufB, n);
    scatter_edges<128><<<cdiv((long long)E * 32, T), T, 0, stream>>>(src, dst, dinv, bufA, bufB, E);

    // --- layer 3: h3 = relu(agg2) @ W3 ; agg3 (no relu) ---------------------
    gemm_wmma_f32<128, 64, true><<<blk64, 256, 0, stream>>>(bufB, W3, bufA);
    self_bias<64><<<cdiv((long long)n * 16, T), T, 0, stream>>>(bufA, dinv, b3, bufB, n);
    scatter_edges<64><<<cdiv((long long)E * 16, T), T, 0, stream>>>(src, dst, dinv, bufA, bufB, E);

    // --- global mean pool ---------------------------------------------------
    fill_f32<<<cdiv(GCN_N_GRAPHS * 64 + GCN_N_GRAPHS, T), T, 0, stream>>>(
        sums, 0.0f, GCN_N_GRAPHS * 64 + GCN_N_GRAPHS);
    pool_sum<<<cdiv(n, 256), 64, 0, stream>>>(bufB, bt, sums, cnt, n);
    pool_div<<<cdiv(out_size, T), T, 0, stream>>>(sums, cnt, outf, out_size);
}